// transferFunctionModel_44933947851110
// MI455X (gfx1250) — compile-verified
//
#include <hip/hip_runtime.h>

// Elementwise transfer-function model, MI455X (gfx1250, wave32).
// Strategy: shared-log factorization of all pow() calls -> v_log_f32/v_exp_f32,
// x^-0.25 -> v_rsq(v_sqrt), divides -> v_rcp, sin -> v_sin (period-1 input),
// 4 rows/thread streamed via b128 non-temporal loads/stores.

typedef __attribute__((ext_vector_type(4))) float v4f;

__device__ __forceinline__ float tf_one(float ommh2, float ombh2, float k,
                                        const float* __restrict__ a,
                                        const float* __restrict__ c) {
  const float L2E      = 1.44269504088896340736f;   // log2(e)
  const float INV2PI   = 0.15915494309189533577f;   // 1/(2*pi)  (v_sin period = 1.0)
  const float LOG2_10P4 = 3.37851162325372979f;     // log2(10.4)

  // x = k / (ommh2 - ombh2)
  float x  = k * __builtin_amdgcn_rcpf(ommh2 - ombh2);
  float lx = __builtin_amdgcn_logf(x);              // log2(x), one log feeds 4 powers

  float p1 = 59.0998f * __builtin_amdgcn_exp2f(1.49177f * lx);
  float p2 = 4658.01f * __builtin_amdgcn_exp2f(4.02755f * lx);
  float p3 = 3170.79f * __builtin_amdgcn_exp2f(6.06f    * lx);
  float p4 = 150.089f * __builtin_amdgcn_exp2f(7.28478f * lx);
  float s  = 1.0f + p1 + p2 + p3 + p4;
  // s^(-0.25) = rsq(sqrt(s)) : 2 hw ops instead of log+exp
  float T_nw = __builtin_amdgcn_rsqf(__builtin_amdgcn_sqrtf(s));

  float lb = __builtin_amdgcn_logf(ombh2);          // log2(ombh2): feeds 5 powers
  float lm = __builtin_amdgcn_logf(ommh2);          // log2(ommh2): feeds 7 powers

  float f_alpha = a[7]  - a[8]  * __builtin_amdgcn_exp2f(a[9]  * lb)
                        + a[10] * __builtin_amdgcn_exp2f(a[11] * lm);
  float f_beta  = a[12] - a[13] * __builtin_amdgcn_exp2f(a[14] * lb)
                        + a[15] * __builtin_amdgcn_exp2f(a[16] * lm);
  float f_node  = a[17] * __builtin_amdgcn_exp2f(a[18] * lm);

  float den = c[0] * __builtin_amdgcn_exp2f(c[1] * lb)
            + c[2] * __builtin_amdgcn_exp2f(c[3] * lm)
            + c[4] * __builtin_amdgcn_exp2f(c[5] * lb + c[6] * lm); // fused b^c5 * m^c6
  float s_GA = __builtin_amdgcn_rcpf(den);

  // k_Silk = 1.6 * ombh2^0.52 * ommh2^0.73 * (1 + (10.4*ommh2)^-0.95)
  float silk_main = 1.6f * __builtin_amdgcn_exp2f(0.52f * lb + 0.73f * lm);
  float silk_corr = 1.0f + __builtin_amdgcn_exp2f(-0.95f * (LOG2_10P4 + lm));
  float k_Silk    = silk_main * silk_corr;

  float ksga  = k * s_GA;
  float lksga = __builtin_amdgcn_logf(ksga);

  // f_amp = f_alpha / (a0 + (f_beta/ksga)^a1)
  float lfb   = __builtin_amdgcn_logf(f_beta);
  float f_amp = f_alpha * __builtin_amdgcn_rcpf(
                    a[0] + __builtin_amdgcn_exp2f(a[1] * (lfb - lksga)));

  // f_Silk = (k / k_Silk)^a2
  float lk     = __builtin_amdgcn_logf(k);
  float lkS    = __builtin_amdgcn_logf(k_Silk);
  float f_Silk = __builtin_amdgcn_exp2f(a[2] * (lk - lkS));

  // f_osc = a3*ksga / (a4 + f_node/ksga^a5)^a6
  float oscden = a[4] + f_node * __builtin_amdgcn_exp2f(-a[5] * lksga);
  float f_osc  = a[3] * ksga *
                 __builtin_amdgcn_exp2f(-a[6] * __builtin_amdgcn_logf(oscden));

  float T_w = 1.0f + f_amp
                   * __builtin_amdgcn_exp2f(-L2E * f_Silk)          // exp(-f_Silk)
                   * __builtin_amdgcn_sinf(f_osc * INV2PI);         // sin(f_osc)
  return T_nw * T_w;
}

__global__ __launch_bounds__(256) void transferFunctionModel_kernel(
    const float* __restrict__ in_data,   // N*3 interleaved [ommh2, ombh2, k]
    const float* __restrict__ params,    // 26 broadcast scalars
    float* __restrict__ out, int n) {
  // Broadcast params: uniform addresses + restrict -> s_load into SGPRs.
  float a[19], c[7];
#pragma unroll
  for (int i = 0; i < 19; ++i) a[i] = params[i];
#pragma unroll
  for (int i = 0; i < 7; ++i)  c[i] = params[19 + i];

  int tid  = blockIdx.x * blockDim.x + threadIdx.x;
  int base = tid * 4;
  if (base >= n) return;

  if (base + 4 <= n) {
    // 4 rows = 48 B per thread: three b128 NT loads, fully coalesced.
    const v4f* inv = reinterpret_cast<const v4f*>(in_data);
    v4f v0 = __builtin_nontemporal_load(&inv[tid * 3 + 0]);
    v4f v1 = __builtin_nontemporal_load(&inv[tid * 3 + 1]);
    v4f v2 = __builtin_nontemporal_load(&inv[tid * 3 + 2]);

    float om[4] = {v0.x, v0.w, v1.z, v2.y};
    float ob[4] = {v0.y, v1.x, v1.w, v2.z};
    float kk[4] = {v0.z, v1.y, v2.x, v2.w};

    v4f r;
    r.x = tf_one(om[0], ob[0], kk[0], a, c);
    r.y = tf_one(om[1], ob[1], kk[1], a, c);
    r.z = tf_one(om[2], ob[2], kk[2], a, c);
    r.w = tf_one(om[3], ob[3], kk[3], a, c);

    __builtin_nontemporal_store(r, reinterpret_cast<v4f*>(out) + tid);
  } else {
    // Scalar tail (not reached when n % 4 == 0, but kept for safety).
    for (int i = base; i < n; ++i) {
      float om = in_data[3 * i + 0];
      float ob = in_data[3 * i + 1];
      float kk = in_data[3 * i + 2];
      out[i] = tf_one(om, ob, kk, a, c);
    }
  }
}

extern "C" void kernel_launch(void* const* d_in, const int* in_sizes, int n_in,
                              void* d_out, int out_size, void* d_ws, size_t ws_size,
                              hipStream_t stream) {
  (void)in_sizes; (void)n_in; (void)d_ws; (void)ws_size;
  const float* in_data = (const float*)d_in[0];
  const float* params  = (const float*)d_in[1];
  float* out = (float*)d_out;

  int n = out_size;                       // one output per row
  int threads_needed = (n + 3) / 4;       // 4 rows per thread
  int blocks = (threads_needed + 255) / 256;
  transferFunctionModel_kernel<<<blocks, 256, 0, stream>>>(in_data, params, out, n);
}